// Attn_28690381538079
// MI455X (gfx1250) — compile-verified
//
#include <hip/hip_runtime.h>

#define S_LEN 2048
#define B_SZ  64
#define D_SZ  1024

typedef __attribute__((ext_vector_type(2))) float v2f;
typedef __attribute__((ext_vector_type(4))) float v4f;
typedef __attribute__((ext_vector_type(8))) float v8f;

// ---------------------------------------------------------------------------
// Kernel 1: v = hidden @ W   (M=64, K=1024, N=1024), fp32 WMMA 16x16x4.
// One wave (32 lanes) per 16x16 output tile; 4 M-tiles * 64 N-tiles = 256 waves.
// Fragment layouts per CDNA5 ISA 7.12.2:
//   A (16x4 f32):  lane L -> M = L%16 ; VGPR j holds K = k0 + 2*(L/16) + j
//   B (4x16 f32):  lane L -> N = L%16 ; VGPR j holds K = k0 + 2*(L/16) + j
//   C/D (16x16):   lane L, VGPR r -> M = r + 8*(L/16), N = L%16
// ---------------------------------------------------------------------------
__global__ __launch_bounds__(32) void proj_hw_wmma(const float* __restrict__ H,
                                                   const float* __restrict__ W,
                                                   float* __restrict__ V) {
  const int tile = blockIdx.x;      // 0..255
  const int mt   = tile >> 6;       // 0..3
  const int nt   = tile & 63;       // 0..63
  const int lane = threadIdx.x;     // 0..31
  const int half = lane >> 4;       // 0 or 1
  const int l16  = lane & 15;

  const int m  = mt * 16 + l16;     // A row this lane supplies
  const int n0 = nt * 16;

  const float* __restrict__ hrow = H + (size_t)m * D_SZ;

  v8f acc = {};
  for (int k0 = 0; k0 < D_SZ; k0 += 4) {
    const int ka = k0 + 2 * half;
    // A fragment: two consecutive K values -> one 8B load
    v2f a = *reinterpret_cast<const v2f*>(hrow + ka);
    // B fragment: two rows of W (stride D), same column n0+l16
    v2f b;
    b.x = W[(size_t)ka * D_SZ + n0 + l16];
    b.y = W[(size_t)(ka + 1) * D_SZ + n0 + l16];

    acc = __builtin_amdgcn_wmma_f32_16x16x4_f32(
        /*neg_a=*/false, a, /*neg_b=*/false, b,
        /*c_mod=*/(short)0, acc, /*reuse_a=*/false, /*reuse_b=*/false);
  }

#pragma unroll
  for (int r = 0; r < 8; ++r) {
    const int mm = mt * 16 + r + 8 * half;
    V[(size_t)mm * D_SZ + n0 + l16] = acc[r];
  }
}

// ---------------------------------------------------------------------------
// Kernel 2: energies[b,s] = v[b,:] . enc[s,b,:]
// One wave per (s,b) row. enc row is 4 KB contiguous -> 16B-vector coalesced,
// non-temporal (single-use 512 MB stream). v row (4 KB per b) stays cached.
// ---------------------------------------------------------------------------
__global__ __launch_bounds__(256) void energies_dot(const float* __restrict__ enc,
                                                    const float* __restrict__ V,
                                                    float* __restrict__ E) {
  const int wavesPerBlock = blockDim.x >> 5;
  const int row = blockIdx.x * wavesPerBlock + (threadIdx.x >> 5);  // = s*B + b
  if (row >= S_LEN * B_SZ) return;
  const int lane = threadIdx.x & 31;
  const int b = row & (B_SZ - 1);
  const int s = row >> 6;

  const v4f* __restrict__ ep = reinterpret_cast<const v4f*>(enc + (size_t)row * D_SZ);
  const v4f* __restrict__ vp = reinterpret_cast<const v4f*>(V + (size_t)b * D_SZ);

  float a = 0.0f;
#pragma unroll
  for (int t = 0; t < D_SZ / (4 * 32); ++t) {  // 8 iterations
    const v4f e  = __builtin_nontemporal_load(&ep[t * 32 + lane]);
    const v4f vv = vp[t * 32 + lane];
    a = fmaf(e.x, vv.x, a);
    a = fmaf(e.y, vv.y, a);
    a = fmaf(e.z, vv.z, a);
    a = fmaf(e.w, vv.w, a);
  }
#pragma unroll
  for (int off = 16; off > 0; off >>= 1) a += __shfl_xor(a, off, 32);
  if (lane == 0) E[(size_t)b * S_LEN + s] = a;
}

// ---------------------------------------------------------------------------
// Kernel 3: out[b,0,:] = softmax(E[b,:])  over S=2048. One block per b,
// 256 threads x 8 elements, stable softmax with LDS cross-wave reduction.
// (Bias term hidden.b_attn is a per-b constant -> softmax-invariant -> dropped.)
// ---------------------------------------------------------------------------
__global__ __launch_bounds__(256) void softmax_row(const float* __restrict__ E,
                                                   float* __restrict__ out) {
  const int b    = blockIdx.x;
  const int tid  = threadIdx.x;
  const int lane = tid & 31;
  const int wid  = tid >> 5;
  __shared__ float red[8];
  __shared__ float bcast;

  const float* __restrict__ row = E + (size_t)b * S_LEN;
  float vals[8];
  float m = -3.402823466e38f;
#pragma unroll
  for (int i = 0; i < 8; ++i) {
    vals[i] = row[i * 256 + tid];
    m = fmaxf(m, vals[i]);
  }
#pragma unroll
  for (int off = 16; off > 0; off >>= 1) m = fmaxf(m, __shfl_xor(m, off, 32));
  if (lane == 0) red[wid] = m;
  __syncthreads();
  if (tid == 0) {
    float t = red[0];
#pragma unroll
    for (int i = 1; i < 8; ++i) t = fmaxf(t, red[i]);
    bcast = t;
  }
  __syncthreads();
  m = bcast;
  __syncthreads();

  float sum = 0.0f;
#pragma unroll
  for (int i = 0; i < 8; ++i) {
    vals[i] = __expf(vals[i] - m);
    sum += vals[i];
  }
#pragma unroll
  for (int off = 16; off > 0; off >>= 1) sum += __shfl_xor(sum, off, 32);
  if (lane == 0) red[wid] = sum;
  __syncthreads();
  if (tid == 0) {
    float t = 0.0f;
#pragma unroll
    for (int i = 0; i < 8; ++i) t += red[i];
    bcast = t;
  }
  __syncthreads();
  const float inv = 1.0f / bcast;
#pragma unroll
  for (int i = 0; i < 8; ++i) out[(size_t)b * S_LEN + i * 256 + tid] = vals[i] * inv;
}

// ---------------------------------------------------------------------------
extern "C" void kernel_launch(void* const* d_in, const int* in_sizes, int n_in,
                              void* d_out, int out_size, void* d_ws, size_t ws_size,
                              hipStream_t stream) {
  const float* hidden = (const float*)d_in[0];  // [1,B,D]
  const float* enc    = (const float*)d_in[1];  // [S,B,D]
  const float* W      = (const float*)d_in[2];  // [D,D]
  // d_in[3] = b_attn: softmax-invariant per-row constant -> not needed.

  float* V = (float*)d_ws;                                    // [B,D]  256 KB
  float* E = (float*)((char*)d_ws + (size_t)B_SZ * D_SZ * 4); // [B,S]  512 KB
  float* out = (float*)d_out;                                 // [B,1,S]

  // 1) v = hidden @ W   (256 tiles, one wave each)
  proj_hw_wmma<<<256, 32, 0, stream>>>(hidden, W, V);

  // 2) energies: S*B waves, 8 waves per 256-thread block
  const int rows = S_LEN * B_SZ;
  energies_dot<<<rows / 8, 256, 0, stream>>>(enc, V, E);

  // 3) softmax per batch row
  softmax_row<<<B_SZ, 256, 0, stream>>>(E, out);
}